// DIGRAC_Unroll_Sync_30709016166868
// MI455X (gfx1250) — compile-verified
//
#include <hip/hip_runtime.h>

// DIGRAC-style pipeline for MI455X (gfx1250, wave32, WMMA).
// Phases: [zero ws] -> [MLP GEMMs via V_WMMA_F32_16X16X4_F32] -> [scatter A_sk dense]
// -> [compact nonzeros -> sparse H (cos/sin)] -> [DIMPA sparse hops] -> [score]
// -> [20x (y=e^{i*s}; acc=ALPHA*y + H@y; s=angle(acc) mod 2pi)].

#define NN 4096
#define FD 256
#define HIDD 32
#define NE 131072
#define NSTEPS 20
#define ALPHA_C 0.01f
#define TWO_PI_C 6.283185307179586f

typedef float v2f __attribute__((ext_vector_type(2)));
typedef float v4f __attribute__((ext_vector_type(4)));
typedef float v8f __attribute__((ext_vector_type(8)));

__device__ __forceinline__ v8f wmma_f32x4(v2f a, v2f b, v8f c) {
  // 8 args: (neg_a, A, neg_b, B, c_mod, C, reuse_a, reuse_b)
  return __builtin_amdgcn_wmma_f32_16x16x4_f32(false, a, false, b, (short)0, c,
                                               false, false);
}

// ---------------------------------------------------------------------------
// MLP: out[16 rows x 32] = relu(X_tile @ W0) @ W1, one wave per 16-row tile.
// A-frag layout (16x4 f32): lane = 16*(k>=2) + m, vgpr = k%2.
// B-frag layout (4x16 f32): lane = 16*(k>=2) + n, vgpr = k%2.
// C/D layout (16x16 f32):   lane = n + 16*(m>=8), vgpr = m%8.
// ---------------------------------------------------------------------------
__global__ __launch_bounds__(32) void mlp16_kernel(const float* __restrict__ X,
                                                   const float* __restrict__ W0,
                                                   const float* __restrict__ W1,
                                                   float* __restrict__ out) {
  __shared__ float xt[16 * FD];    // 16 KB row-major X tile
  __shared__ float ht[16 * HIDD];  // 2 KB intermediate (relu(X@W0))
  const int lane = threadIdx.x;
  const int half = lane >> 4;   // K sub-block select
  const int m16  = lane & 15;   // row (A) / col (B,D)
  const int row0 = blockIdx.x * 16;

  // coalesced 16 KB contiguous copy: global -> LDS
  {
    const v4f* s = (const v4f*)(X + (size_t)row0 * FD);
    v4f* d = (v4f*)xt;
    #pragma unroll
    for (int i = lane; i < 16 * FD / 4; i += 32) d[i] = s[i];
  }
  __syncthreads();

  v8f zero = {};
  v8f acc0 = zero, acc1 = zero;  // N-tiles [0..15], [16..31]

  #pragma unroll 8
  for (int kb = 0; kb < FD; kb += 4) {
    const int kA = kb + 2 * half;
    v2f a = *(const v2f*)&xt[m16 * FD + kA];
    v2f b0, b1;
    b0.x = W0[kA * HIDD + m16];
    b0.y = W0[(kA + 1) * HIDD + m16];
    b1.x = W0[kA * HIDD + 16 + m16];
    b1.y = W0[(kA + 1) * HIDD + 16 + m16];
    acc0 = wmma_f32x4(a, b0, acc0);
    acc1 = wmma_f32x4(a, b1, acc1);
  }

  // relu, park D-layout back as row-major 16x32 in LDS
  #pragma unroll
  for (int r = 0; r < 8; ++r) {
    const int m = r + 8 * half;
    ht[m * HIDD + m16]      = fmaxf(acc0[r], 0.0f);
    ht[m * HIDD + 16 + m16] = fmaxf(acc1[r], 0.0f);
  }
  __syncthreads();

  acc0 = zero; acc1 = zero;
  #pragma unroll
  for (int kb = 0; kb < HIDD; kb += 4) {
    const int kA = kb + 2 * half;
    v2f a = *(const v2f*)&ht[m16 * HIDD + kA];
    v2f b0, b1;
    b0.x = W1[kA * HIDD + m16];
    b0.y = W1[(kA + 1) * HIDD + m16];
    b1.x = W1[kA * HIDD + 16 + m16];
    b1.y = W1[(kA + 1) * HIDD + 16 + m16];
    acc0 = wmma_f32x4(a, b0, acc0);
    acc1 = wmma_f32x4(a, b1, acc1);
  }

  #pragma unroll
  for (int r = 0; r < 8; ++r) {
    const int m = r + 8 * half;
    out[(size_t)(row0 + m) * HIDD + m16]      = acc0[r];
    out[(size_t)(row0 + m) * HIDD + 16 + m16] = acc1[r];
  }
}

// ---------------------------------------------------------------------------
__global__ void zero_kernel(float* __restrict__ p, size_t n) {
  size_t i = (size_t)blockIdx.x * blockDim.x + threadIdx.x;
  const size_t stride = (size_t)gridDim.x * blockDim.x;
  for (; i < n; i += stride) p[i] = 0.0f;
}

// A_sk[s,d] += w ; A_sk[d,s] -= w  (duplicates accumulate like jnp .at[].add)
__global__ void build_ask_kernel(const int* __restrict__ src,
                                 const int* __restrict__ dst,
                                 const float* __restrict__ w,
                                 float* __restrict__ A) {
  const int e = blockIdx.x * blockDim.x + threadIdx.x;
  if (e >= NE) return;
  const int s = src[e], d = dst[e];
  const float we = w[e];
  atomicAdd(&A[(size_t)s * NN + d], we);
  atomicAdd(&A[(size_t)d * NN + s], -we);
}

// compact nonzeros of dense A_sk into sparse H with precomputed cos/sin.
__global__ void compact_kernel(const float* __restrict__ A,
                               int* __restrict__ rows, int* __restrict__ cols,
                               float* __restrict__ hre, float* __restrict__ him,
                               int* __restrict__ counter) {
  size_t i = (size_t)blockIdx.x * blockDim.x + threadIdx.x;
  const size_t stride = (size_t)gridDim.x * blockDim.x;
  const size_t total = (size_t)NN * NN;
  for (; i < total; i += stride) {
    const float a = A[i];
    if (a != 0.0f) {
      const int p = atomicAdd(counter, 1);
      rows[p] = (int)(i >> 12);
      cols[p] = (int)(i & (NN - 1));
      float s, c;
      sincosf(a, &s, &c);
      hre[p] = c;
      him[p] = s;
    }
  }
}

// edge-parallel SpMM: out[ia[e], :] += w[e] * x[ib[e], :]; 8 threads/edge.
__global__ void spmm_kernel(const int* __restrict__ ia, const int* __restrict__ ib,
                            const float* __restrict__ w, const float* __restrict__ x,
                            float* __restrict__ out) {
  const int t = blockIdx.x * blockDim.x + threadIdx.x;
  const int e = t >> 3;
  if (e >= NE) return;
  const int c4 = (t & 7) * 4;
  const int a = ia[e], b = ib[e];
  const float we = w[e];
  const float* xb = x + (size_t)b * HIDD + c4;
  float* ob = out + (size_t)a * HIDD + c4;
  const v4f v = *(const v4f*)xb;
  atomicAdd(&ob[0], we * v.x);
  atomicAdd(&ob[1], we * v.y);
  atomicAdd(&ob[2], we * v.z);
  atomicAdd(&ob[3], we * v.w);
}

// z = [ws0*xs+ws1*h1s+ws2*h2s, wt0*xt+wt1*h1t+wt2*h2t]; score = 2pi*sigmoid(z.lw + b)
__global__ void combine_score_kernel(const float* __restrict__ xs,
                                     const float* __restrict__ h1s,
                                     const float* __restrict__ h2s,
                                     const float* __restrict__ xt,
                                     const float* __restrict__ h1t,
                                     const float* __restrict__ h2t,
                                     const float* __restrict__ ws,
                                     const float* __restrict__ wt,
                                     const float* __restrict__ lw,
                                     const float* __restrict__ lb,
                                     float* __restrict__ score) {
  const int i = blockIdx.x * blockDim.x + threadIdx.x;
  if (i >= NN) return;
  const float ws0 = ws[0], ws1 = ws[1], ws2 = ws[2];
  const float wt0 = wt[0], wt1 = wt[1], wt2 = wt[2];
  const size_t o = (size_t)i * HIDD;
  float dot = lb[0];
  #pragma unroll 8
  for (int c = 0; c < HIDD; ++c)
    dot += (ws0 * xs[o + c] + ws1 * h1s[o + c] + ws2 * h2s[o + c]) * lw[c];
  #pragma unroll 8
  for (int c = 0; c < HIDD; ++c)
    dot += (wt0 * xt[o + c] + wt1 * h1t[o + c] + wt2 * h2t[o + c]) * lw[HIDD + c];
  const float sig = 1.0f / (1.0f + expf(-dot));
  score[i] = TWO_PI_C * sig;
}

// y = e^{i*score};  acc initialized to ALPHA*y
__global__ void y_init_kernel(const float* __restrict__ score,
                              float* __restrict__ y, float* __restrict__ acc) {
  const int i = blockIdx.x * blockDim.x + threadIdx.x;
  if (i >= NN) return;
  float s, c;
  sincosf(score[i], &s, &c);
  y[2 * i] = c;
  y[2 * i + 1] = s;
  acc[2 * i] = ALPHA_C * c;
  acc[2 * i + 1] = ALPHA_C * s;
}

// acc[i] += H[i,j] * y[j] over sparse nonzeros (complex)
__global__ void hmatvec_kernel(const int* __restrict__ rows,
                               const int* __restrict__ cols,
                               const float* __restrict__ hre,
                               const float* __restrict__ him,
                               const float* __restrict__ y,
                               float* __restrict__ acc,
                               const int* __restrict__ counter) {
  const int e = blockIdx.x * blockDim.x + threadIdx.x;
  if (e >= *counter) return;
  const int i = rows[e], j = cols[e];
  const float yr = y[2 * j], yi = y[2 * j + 1];
  const float hr = hre[e], hi = him[e];
  atomicAdd(&acc[2 * i], hr * yr - hi * yi);
  atomicAdd(&acc[2 * i + 1], hr * yi + hi * yr);
}

// score = angle(acc) mod 2pi
__global__ void angle_kernel(const float* __restrict__ acc,
                             float* __restrict__ score) {
  const int i = blockIdx.x * blockDim.x + threadIdx.x;
  if (i >= NN) return;
  float a = atan2f(acc[2 * i + 1], acc[2 * i]);
  if (a < 0.0f) a += TWO_PI_C;
  score[i] = a;
}

// ---------------------------------------------------------------------------
extern "C" void kernel_launch(void* const* d_in, const int* in_sizes, int n_in,
                              void* d_out, int out_size, void* d_ws, size_t ws_size,
                              hipStream_t stream) {
  (void)in_sizes; (void)n_in; (void)out_size; (void)ws_size;

  const int* edge_index = (const int*)d_in[0];
  const int* src = edge_index;            // edge_index[0]
  const int* dst = edge_index + NE;       // edge_index[1]
  const float* ew    = (const float*)d_in[1];
  const float* feats = (const float*)d_in[2];
  const float* w_s0  = (const float*)d_in[3];
  const float* w_s1  = (const float*)d_in[4];
  const float* w_t0  = (const float*)d_in[5];
  const float* w_t1  = (const float*)d_in[6];
  const float* dws   = (const float*)d_in[7];
  const float* dwt   = (const float*)d_in[8];
  const float* lw    = (const float*)d_in[9];
  const float* lb    = (const float*)d_in[10];
  float* score = (float*)d_out;           // [N,1] fp32

  // workspace carve-up (floats). Zeroed region is contiguous at the front.
  float* ws = (float*)d_ws;
  size_t off = 0;
  float* A = ws + off;       off += (size_t)NN * NN;   // 64 MB dense A_sk
  int* counter = (int*)(ws + off); off += 64;          // nnz counter (padded)
  float* h1s = ws + off;     off += (size_t)NN * HIDD;
  float* h2s = ws + off;     off += (size_t)NN * HIDD;
  float* h1t = ws + off;     off += (size_t)NN * HIDD;
  float* h2t = ws + off;     off += (size_t)NN * HIDD;
  const size_t zero_words = off;                       // everything above needs 0s
  float* xs  = ws + off;     off += (size_t)NN * HIDD;
  float* xt  = ws + off;     off += (size_t)NN * HIDD;
  int* rows  = (int*)(ws + off); off += 2 * (size_t)NE;  // nnz <= 2E
  int* cols  = (int*)(ws + off); off += 2 * (size_t)NE;
  float* hre = ws + off;     off += 2 * (size_t)NE;
  float* him = ws + off;     off += 2 * (size_t)NE;
  float* yv  = ws + off;     off += 2 * (size_t)NN;
  float* acc = ws + off;     off += 2 * (size_t)NN;

  // 1) zero A_sk + counter + hop accumulators
  zero_kernel<<<2048, 256, 0, stream>>>(ws, zero_words);

  // 2) feature MLPs (WMMA f32), one wave per 16-row tile
  mlp16_kernel<<<NN / 16, 32, 0, stream>>>(feats, w_s0, w_s1, xs);
  mlp16_kernel<<<NN / 16, 32, 0, stream>>>(feats, w_t0, w_t1, xt);

  // 3) dense skew-symmetric A_sk via scatter-atomics
  build_ask_kernel<<<NE / 256, 256, 0, stream>>>(src, dst, ew, A);

  // 4) compact nonzeros -> sparse H (cos/sin precomputed); A_sk is L2-resident
  compact_kernel<<<4096, 256, 0, stream>>>(A, rows, cols, hre, him, counter);

  // 5) DIMPA hops: spmm (src<-dst) for s, spmm_t (dst<-src) for t
  const int spmm_blocks = (NE * 8) / 256;
  spmm_kernel<<<spmm_blocks, 256, 0, stream>>>(src, dst, ew, xs, h1s);
  spmm_kernel<<<spmm_blocks, 256, 0, stream>>>(src, dst, ew, h1s, h2s);
  spmm_kernel<<<spmm_blocks, 256, 0, stream>>>(dst, src, ew, xt, h1t);
  spmm_kernel<<<spmm_blocks, 256, 0, stream>>>(dst, src, ew, h1t, h2t);

  // 6) initial angles
  combine_score_kernel<<<NN / 256, 256, 0, stream>>>(xs, h1s, h2s, xt, h1t, h2t,
                                                     dws, dwt, lw, lb, score);

  // 7) 20 projected power iterations (sparse complex matvec, ~4 MB/iter)
  for (int it = 0; it < NSTEPS; ++it) {
    y_init_kernel<<<NN / 256, 256, 0, stream>>>(score, yv, acc);
    hmatvec_kernel<<<(2 * NE) / 256, 256, 0, stream>>>(rows, cols, hre, him, yv,
                                                       acc, counter);
    angle_kernel<<<NN / 256, 256, 0, stream>>>(acc, score);
  }
}